// AudioFeatureExtractor_12128987644570
// MI455X (gfx1250) — compile-verified
//
#include <hip/hip_runtime.h>
#include <math.h>

typedef __attribute__((ext_vector_type(2))) float v2f;
typedef __attribute__((ext_vector_type(4))) float v4f;
typedef __attribute__((ext_vector_type(8))) float v8f;

#define N_FFT   2048
#define HOP     512
#define NTILE_K 64              // bins 0..1023; Nyquist (1024) handled by VALU
#define KSTEPS  512             // 2048 / 4 (K per f32 WMMA)
#define SPAN    (15*HOP + N_FFT)        // 9728 samples per 16-frame tile
#define LDS_RAW_SZ (SPAN + (SPAN>>7) + 4)

// ---------------------------------------------------------------------------
// Kernel 1: Hann-windowed DFT basis pre-packed into 16x16x4 WMMA B-operand
// lane layout.  float4 {cos_x, cos_y, -sin_x, -sin_y} per (tile, kc, lane).
//   lane<16 : elems for K=k0+0 / k0+1, N = lane
//   lane>=16: elems for K=k0+2 / k0+3, N = lane-16
// ---------------------------------------------------------------------------
__global__ void build_basis(float* __restrict__ basis) {
    int t = blockIdx.x * blockDim.x + threadIdx.x;
    const int total = NTILE_K * KSTEPS * 32;
    if (t >= total) return;
    int lane = t & 31;
    int kc   = (t >> 5) & (KSTEPS - 1);
    int tile = t >> (5 + 9);
    int k  = tile * 16 + (lane & 15);           // 0..1023, always valid
    int n0 = kc * 4 + ((lane >> 4) << 1);
    const float twopi = 6.28318530717958647692f;
    float wn0 = 0.5f - 0.5f * cosf(twopi * (float)n0     * (1.0f/2048.0f));
    float wn1 = 0.5f - 0.5f * cosf(twopi * (float)(n0+1) * (1.0f/2048.0f));
    int ph0 = (n0       * k) & 2047;            // exact modular phase reduction
    int ph1 = ((n0 + 1) * k) & 2047;
    float s0, c0, s1, c1;
    sincosf((float)ph0 * (twopi/2048.0f), &s0, &c0);
    sincosf((float)ph1 * (twopi/2048.0f), &s1, &c1);
    v4f o = { wn0 * c0, wn1 * c1, -wn0 * s0, -wn1 * s1 };
    ((v4f*)basis)[t] = o;
}

// ---------------------------------------------------------------------------
// Kernel 2: one workgroup (256 thr = 8 waves) per (channel, 16-frame tile).
// Each wave runs 2 passes of 4 bin-tiles: per K-step 1 A-fetch feeds 8 WMMAs.
// ---------------------------------------------------------------------------
__global__ void __launch_bounds__(256)
audio_feat_kernel(const float* __restrict__ x,      // [T, 4] interleaved
                  const float* __restrict__ basis,  // packed WMMA-B basis
                  float* __restrict__ out,          // [4, F, 4]
                  int T, int F, int tilesPerCh)
{
    __shared__ float raw[LDS_RAW_SZ];   // skewed: word(i) = i + (i>>7)
    __shared__ float facc[5][16];       // rms_p, log_sp, sp, S, f*S per frame
    __shared__ float zc[16];
    __shared__ float nyq[16];

    const int tid  = threadIdx.x;
    const int lane = tid & 31;
    const int wave = tid >> 5;
    const int bx   = blockIdx.x;
    const int c    = bx / tilesPerCh;
    const int tileF = bx - c * tilesPerCh;
    const long f0   = (long)tileF * 16;
    const long base = f0 * HOP;                 // start in zero-padded coords

    // ---- stage raw (zero-padded) span into skewed LDS ----
    for (int i = tid; i < SPAN; i += 256) {
        long tpos = base + i - (N_FFT / 2);
        float v = 0.0f;
        if (tpos >= 0 && tpos < (long)T) v = x[tpos * 4 + c];
        raw[i + (i >> 7)] = v;
    }
    if (tid < 80) ((float*)facc)[tid] = 0.0f;
    if (tid < 16) { zc[tid] = 0.0f; nyq[tid] = 0.0f; }
    __syncthreads();

    // ---- WMMA DFT: D = frames(16xK) * basis(Kx16), f32 16x16x4 ----
    const int mloc  = lane & 15;
    const int abase = mloc * HOP + ((lane >> 4) << 1);  // A-operand lane layout
    const int mrow  = (lane >> 4) * 8;                  // row M offset per half
    for (int g0 = wave * 8; g0 < wave * 8 + 8; g0 += 4) {   // wave-uniform
        v8f accC[4] = {};
        v8f accS[4] = {};
        const v4f* bp = (const v4f*)basis + ((size_t)g0 * KSTEPS) * 32 + lane;
        int ai = abase;
        #pragma unroll 2
        for (int kc = 0; kc < KSTEPS; ++kc, bp += 32, ai += 4) {
            v2f a;
            a.x = raw[ai + (ai >> 7)];
            int ai1 = ai + 1;
            a.y = raw[ai1 + (ai1 >> 7)];
            #pragma unroll
            for (int t = 0; t < 4; ++t) {
                v4f b = bp[(size_t)t * (KSTEPS * 32)];  // 256KB imm offsets
                v2f bc = {b.x, b.y};
                v2f bs = {b.z, b.w};
                accC[t] = __builtin_amdgcn_wmma_f32_16x16x4_f32(
                    false, a, false, bc, (short)0, accC[t], false, false);
                accS[t] = __builtin_amdgcn_wmma_f32_16x16x4_f32(
                    false, a, false, bs, (short)0, accS[t], false, false);
            }
        }
        // ---- epilogue: power + feature partial sums (all bins < 1024) ----
        #pragma unroll
        for (int t = 0; t < 4; ++t) {
            const int   kk   = (g0 + t) * 16 + mloc;
            const float freq = (float)kk * (44100.0f / 2048.0f);
            const float wrm  = (kk == 0) ? 0.5f : 1.0f;     // DC halving
            #pragma unroll
            for (int r = 0; r < 8; ++r) {
                float re = accC[t][r], im = accS[t][r];
                float p  = re * re + im * im;
                float S  = sqrtf(p);
                float sp = fmaxf(1e-10f, p);
                float vv[5];
                vv[0] = wrm * p;        // rms power
                vv[1] = logf(sp);       // flatness log-sum
                vv[2] = sp;             // flatness lin-sum
                vv[3] = S;              // centroid denom
                vv[4] = freq * S;       // centroid numer
                #pragma unroll
                for (int q = 0; q < 5; ++q) {
                    float v = vv[q];
                    v += __shfl_xor(v, 1);
                    v += __shfl_xor(v, 2);
                    v += __shfl_xor(v, 4);
                    v += __shfl_xor(v, 8);  // sum over N within 16-lane half
                    if (mloc == 0) atomicAdd(&facc[q][mrow + r], v);
                }
            }
        }
    }

    // ---- Nyquist bin (k=1024): Nq[m] = sum_n w[n]*(-1)^n * frame[m][n] ----
    {
        const int m   = tid >> 4;
        const int sub = tid & 15;
        const int lo  = sub * 128;
        const float twopi = 6.28318530717958647692f;
        float s = 0.0f;
        for (int j = lo; j < lo + 128; ++j) {
            int il = m * HOP + j;
            float w = 0.5f - 0.5f * cosf(twopi * (float)j * (1.0f/2048.0f));
            float v = raw[il + (il >> 7)];
            s += ((j & 1) ? -w : w) * v;
        }
        atomicAdd(&nyq[m], s);
    }

    // ---- zero-crossing rate (edge-padded), VALU scan over LDS span ----
    {
        const float xf = x[c];
        const float xl = x[(long)(T - 1) * 4 + c];
        const int m   = tid >> 4;
        const int sub = tid & 15;
        const int lo  = sub * 128;
        auto sgn = [&](int j) -> bool {
            int il = m * HOP + j;
            long tp = base + il - (N_FFT / 2);
            float v = raw[il + (il >> 7)];
            v = (tp < 0)        ? xf : v;       // edge-pad correction
            v = (tp >= (long)T) ? xl : v;
            return v < -1e-10f;                 // librosa thresholded signbit
        };
        int cnt = 0;
        bool prev = sgn(lo == 0 ? 0 : lo - 1);
        for (int j = (lo == 0 ? 1 : lo); j < lo + 128; ++j) {
            bool cur = sgn(j);
            cnt += (cur != prev);
            prev = cur;
        }
        if (sub == 0) cnt += (sgn(1) != sgn(0));    // cross[0] = cross[1] dup
        atomicAdd(&zc[m], (float)cnt);
    }
    __syncthreads();

    // ---- finalize 16 frames (fold in Nyquist) ----
    if (tid < 16) {
        long f = f0 + tid;
        float nq = nyq[tid];
        float pn = nq * nq;
        float spn = fmaxf(1e-10f, pn);
        float Sn  = sqrtf(pn);
        float sum_rms = facc[0][tid] + 0.5f * pn;
        float sum_log = facc[1][tid] + logf(spn);
        float sum_sp  = facc[2][tid] + spn;
        float sum_S   = facc[3][tid] + Sn;
        float sum_fS  = facc[4][tid] + 1024.0f * (44100.0f / 2048.0f) * Sn;

        float rms  = sqrtf(2.0f * sum_rms * (1.0f / (2048.0f * 2048.0f)));
        float flat = expf(sum_log * (1.0f / 1025.0f)) / (sum_sp * (1.0f / 1025.0f));
        float cent = sum_fS / ((sum_S <= 1.17549435e-38f) ? 1.0f : sum_S);
        float zcr  = zc[tid] * (1.0f / 2048.0f);
        float* o = out + ((size_t)c * F + f) * 4;
        o[0] = rms; o[1] = zcr; o[2] = flat; o[3] = cent;
    }
}

// ---------------------------------------------------------------------------
extern "C" void kernel_launch(void* const* d_in, const int* in_sizes, int n_in,
                              void* d_out, int out_size, void* d_ws, size_t ws_size,
                              hipStream_t stream) {
    const float* x   = (const float*)d_in[0];
    float* out       = (float*)d_out;
    float* basis     = (float*)d_ws;            // 64*512*32*16 B = 16 MB

    const int T = in_sizes[0] / 4;              // 4194304
    const int F = T / HOP;                      // 8192 (last frame dropped)
    const int tilesPerCh = F / 16;              // 512
    const int totalBasis = NTILE_K * KSTEPS * 32;

    hipLaunchKernelGGL(build_basis,
                       dim3((totalBasis + 255) / 256), dim3(256), 0, stream,
                       basis);
    hipLaunchKernelGGL(audio_feat_kernel,
                       dim3(4 * tilesPerCh), dim3(256), 0, stream,
                       x, basis, out, T, F, tilesPerCh);
}